// BaseNeuron_34600256537573
// MI455X (gfx1250) — compile-verified
//
#include <hip/hip_runtime.h>
#include <math.h>

// LIF scan over T=16 for x[B=16, C=64, T=16, H=64, W=64] (fp32).
// Pure streaming workload: 256 MiB in + 256 MiB out, ~23 us floor at 23.3 TB/s.
// Strategy: 1 thread owns 4 consecutive w-elements of one (b,c,h,w) column.
//   - issue all 16 independent global_load_b128 (NT hint) -> max MLP
//   - run the serial T-recurrence entirely in VGPRs
//   - 16 global_store_b128 (NT hint)
// No WMMA: there is no matmul in this op; the roofline is HBM-bound.

typedef __attribute__((ext_vector_type(4))) float v4f;

#ifndef LIF_T
#define LIF_T 16
#endif

__global__ __launch_bounds__(256) void lif_scan_kernel(
    const float* __restrict__ x,
    const float* __restrict__ decay,
    float* __restrict__ out,
    int n_cols4)                       // total vec4 columns = B*C*H*W/4
{
    const int gid = blockIdx.x * blockDim.x + threadIdx.x;
    if (gid >= n_cols4) return;

    // sigmoid(decay) -- uniform across the grid; decay has 1 element.
    const float d  = decay[0];
    const float ds = 1.0f / (1.0f + __expf(-d));

    // Layout: idx = ((b*C + c)*T + t)*HW + hw ; HW = 4096, T*HW = 65536.
    const int bc  = gid >> 10;          // which (b,c):  HW/4 = 1024 vec4 cols each
    const int hw4 = gid & 1023;         // vec4 position within the HxW plane
    const size_t base = (size_t)bc * (size_t)(LIF_T * 4096) + (size_t)hw4 * 4u;

    // Stage 1: all 16 loads in flight (independent -> 16 outstanding B128/lane).
    v4f v[LIF_T];
#pragma unroll
    for (int t = 0; t < LIF_T; ++t) {
        v[t] = __builtin_nontemporal_load(
            (const v4f*)(x + base + (size_t)t * 4096u));
    }

    // Stage 2: serial LIF recurrence in registers.
    // mem = mem_old * ds * (1 - spike) + x_t ; spike = mem > 0.5.
    // (1 - spike) is exactly 0 or 1, so fma(mem*(1-s), ds, x) is bit-exact
    // w.r.t. the reference's product ordering.
    float m0 = 0.f, m1 = 0.f, m2 = 0.f, m3 = 0.f;
    float s0 = 0.f, s1 = 0.f, s2 = 0.f, s3 = 0.f;
#pragma unroll
    for (int t = 0; t < LIF_T; ++t) {
        m0 = __builtin_fmaf(m0 * (1.0f - s0), ds, v[t].x);
        m1 = __builtin_fmaf(m1 * (1.0f - s1), ds, v[t].y);
        m2 = __builtin_fmaf(m2 * (1.0f - s2), ds, v[t].z);
        m3 = __builtin_fmaf(m3 * (1.0f - s3), ds, v[t].w);
        s0 = (m0 > 0.5f) ? 1.0f : 0.0f;
        s1 = (m1 > 0.5f) ? 1.0f : 0.0f;
        s2 = (m2 > 0.5f) ? 1.0f : 0.0f;
        s3 = (m3 > 0.5f) ? 1.0f : 0.0f;
        v[t] = (v4f){s0, s1, s2, s3};   // spikes already in [0,1]; clip is identity
    }

    // Stage 3: 16 NT stores (write-once data; avoid L2 pollution).
#pragma unroll
    for (int t = 0; t < LIF_T; ++t) {
        __builtin_nontemporal_store(v[t], (v4f*)(out + base + (size_t)t * 4096u));
    }
}

extern "C" void kernel_launch(void* const* d_in, const int* in_sizes, int n_in,
                              void* d_out, int out_size, void* d_ws, size_t ws_size,
                              hipStream_t stream) {
    (void)n_in; (void)d_ws; (void)ws_size; (void)out_size;

    const float* x     = (const float*)d_in[0];   // [B,C,T,H,W] fp32
    const float* decay = (const float*)d_in[1];   // [1] fp32
    float*       out   = (float*)d_out;           // [B,C,T,H,W] fp32

    // vec4 columns = total_elements / (T * 4)
    const int n_cols4 = in_sizes[0] / (LIF_T * 4);   // 67,108,864 / 64 = 1,048,576

    const int block = 256;                            // 8 wave32 waves / block
    const int grid  = (n_cols4 + block - 1) / block;  // 4096 blocks

    lif_scan_kernel<<<grid, block, 0, stream>>>(x, decay, out, n_cols4);
}